// ATHP_26388279066955
// MI455X (gfx1250) — compile-verified
//
#include <hip/hip_runtime.h>
#include <hip/hip_bf16.h>

#define B_    4
#define P_    1536
#define M_    4
#define DPHI_ 32
#define DIN_  128
#define K_    20
#define S_    100
#define PM1   (P_-1)

typedef __attribute__((ext_vector_type(2))) float v2f;
typedef __attribute__((ext_vector_type(8))) float v8f;

__device__ __forceinline__ float fast_tanh(float x) {
    x = fminf(15.0f, fmaxf(-15.0f, x));
    float e = __expf(2.0f * x);
    return (e - 1.0f) / (e + 1.0f);
}
__device__ __forceinline__ float softplus1(float z) {
    return (z > 20.0f) ? z : log1pf(__expf(z));
}
__device__ __forceinline__ float softplus10(float x) {
    float z = 10.0f * x;
    return (z > 20.0f) ? x : 0.1f * log1pf(__expf(z));
}
__device__ __forceinline__ float gelu_f(float x) {
    float u = 0.7978845608028654f * (x + 0.044715f * x * x * x);
    return 0.5f * x * (1.0f + fast_tanh(u));
}
__device__ __forceinline__ float wave_sum(float v) {
    for (int off = 16; off > 0; off >>= 1) v += __shfl_xor(v, off, 32);
    return v;
}

// ---------------------------------------------------------------------------
// Stage A: cumulative online-softmax weighted average -> embed (B,P,128)
// One block per (b,m); 8 waves, each scanning a 192-element chunk; lane = d.
// ---------------------------------------------------------------------------
__global__ void __launch_bounds__(256) athp_scan(
    const float* __restrict__ values,
    const float* __restrict__ preatt,
    const float* __restrict__ mask,
    float* __restrict__ embed,
    float* __restrict__ out)
{
    if (blockIdx.x == 0 && threadIdx.x < 2 * B_) out[threadIdx.x] = 0.0f;

    const int b    = blockIdx.x / M_;
    const int m    = blockIdx.x % M_;
    const int tid  = threadIdx.x;
    const int lane = tid & 31;
    const int wv   = tid >> 5;
    const int CH   = P_ / 8;       // 192
    const int q0   = wv * CH;

    __shared__ float lds_mx[8], lds_s[8], lds_acc[256];

    // pass 1: chunk-local (max, sum, acc) aggregate
    float mx = -INFINITY, s = 0.0f, acc = 0.0f;
    for (int q = q0; q < q0 + CH; ++q) {
        float a = preatt[(b * P_ + q) * M_ + m];
        float v = values[(b * P_ + q) * DPHI_ + lane];
        float nm = fmaxf(mx, a);
        float f = __expf(mx - nm);
        float w = __expf(a - nm);
        s   = s * f + w;
        acc = acc * f + w * v;
        mx  = nm;
    }
    lds_acc[wv * 32 + lane] = acc;
    if (lane == 0) { lds_mx[wv] = mx; lds_s[wv] = s; }
    __syncthreads();

    // exclusive prefix over preceding chunks
    float om = -INFINITY, os = 0.0f, oa = 0.0f;
    for (int j = 0; j < wv; ++j) {
        float jm = lds_mx[j], js = lds_s[j], ja = lds_acc[j * 32 + lane];
        float nm = fmaxf(om, jm);
        float f1 = __expf(om - nm), f2 = __expf(jm - nm);
        os = os * f1 + js * f2;
        oa = oa * f1 + ja * f2;
        om = nm;
    }

    // pass 2: rescan with offset, emit normalized cumulative average
    for (int q = q0; q < q0 + CH; ++q) {
        float a = preatt[(b * P_ + q) * M_ + m];
        float v = values[(b * P_ + q) * DPHI_ + lane];
        float nm = fmaxf(om, a);
        float f = __expf(om - nm);
        float w = __expf(a - nm);
        os = os * f + w;
        oa = oa * f + w * v;
        om = nm;
        embed[(size_t)(b * P_ + q) * DIN_ + m * DPHI_ + lane] =
            (oa / os) * mask[b * P_ + q];
    }
}

// ---------------------------------------------------------------------------
// Stage B: start/conv/omega = act(embed @ W + b) * mask  via f32 WMMA.
// Block = 64 rows x 128 cols x 3 matrices; 8 waves x 12 (rowtile,coltile,mat)
// jobs; A tile in LDS with stride 129 (conflict-free 16-row reads).
// ---------------------------------------------------------------------------
#define ASTRIDE 129

__global__ void __launch_bounds__(256) athp_mlp(
    const float* __restrict__ embed,
    const float* __restrict__ mask,
    const float* __restrict__ Ws, const float* __restrict__ bs,
    const float* __restrict__ Wc, const float* __restrict__ bc,
    const float* __restrict__ Wd, const float* __restrict__ bd,
    float* __restrict__ start_o,
    float* __restrict__ conv_o,
    float* __restrict__ omega_o)
{
    const int b    = blockIdx.x / 24;
    const int rt   = blockIdx.x % 24;
    const int p0   = rt * 64;
    const int tid  = threadIdx.x;
    const int lane = tid & 31;
    const int wv   = tid >> 5;

    __shared__ float ldsA[64 * ASTRIDE];
    for (int idx = tid; idx < 64 * DIN_; idx += 256) {
        int r = idx >> 7, d = idx & 127;
        ldsA[r * ASTRIDE + d] = embed[(size_t)(b * P_ + p0 + r) * DIN_ + d];
    }
    __syncthreads();

    const int row  = lane & 15;          // A row / B col within tile
    const int koff = (lane >> 4) * 2;    // K sub-offset per half-wave

    for (int j = wv; j < 96; j += 8) {
        int mat = j >> 5;
        int rem = j & 31;
        int rowtile = rem >> 3;
        int coltile = rem & 7;
        int j0 = coltile * 16;
        const float* W    = (mat == 0) ? Ws : (mat == 1 ? Wc : Wd);
        const float* bias = (mat == 0) ? bs : (mat == 1 ? bc : bd);

        v8f c = {};
        for (int k0 = 0; k0 < DIN_; k0 += 4) {
            int kk = k0 + koff;
            v2f a, bb;
            a.x  = ldsA[(rowtile * 16 + row) * ASTRIDE + kk];
            a.y  = ldsA[(rowtile * 16 + row) * ASTRIDE + kk + 1];
            bb.x = W[kk * DIN_ + j0 + row];
            bb.y = W[(kk + 1) * DIN_ + j0 + row];
            c = __builtin_amdgcn_wmma_f32_16x16x4_f32(
                    false, a, false, bb, (short)0, c, false, false);
        }
        int col = j0 + row;
        float bv = bias[col];
        for (int r = 0; r < 8; ++r) {
            int mrow = r + ((lane >> 4) << 3);
            int p = p0 + rowtile * 16 + mrow;
            if (p < PM1) {
                float x   = c[r] + bv;
                float npm = mask[b * P_ + p + 1];
                size_t oidx = (size_t)(b * PM1 + p) * DIN_ + col;
                if (mat == 0)      start_o[oidx] = gelu_f(x) * npm;
                else if (mat == 1) conv_o[oidx]  = gelu_f(x) * npm;
                else               omega_o[oidx] = softplus10(x) * npm;
            }
        }
    }
}

// ---------------------------------------------------------------------------
// Stage C/D fused: per (b,t) block.
//  - cell_tau (128 x 100) -> LDS [s][d], stride 129
//  - waves 0..6: 7x2 WMMA tiles of out[s,k] = sum_d ct[d,s] W_int[d,k],
//    softplus + reduce -> integral accumulator
//  - wave 7: cell_t dot W_int[:,type] -> log_sum accumulator
// ---------------------------------------------------------------------------
#define CTS 129

__global__ void __launch_bounds__(256) athp_tail(
    const float* __restrict__ start_i,
    const float* __restrict__ conv_i,
    const float* __restrict__ omega_i,
    const float* __restrict__ seq_times,
    const float* __restrict__ mask,
    const float* __restrict__ taus_u,
    const int*   __restrict__ seq_types,
    const float* __restrict__ W_int,
    const float* __restrict__ b_int,
    float* __restrict__ out)
{
    const int b    = blockIdx.x / PM1;
    const int t    = blockIdx.x % PM1;
    const int tid  = threadIdx.x;
    const int lane = tid & 31;
    const int wv   = tid >> 5;

    __shared__ float ct[S_ * CTS];       // 51600 B
    __shared__ float ldsw[DIN_ * K_];    // 10240 B
    __shared__ float sS[DIN_], sC[DIN_], sO[DIN_];
    __shared__ float ltau[S_];
    __shared__ float acc_blk;

    const float tn  = seq_times[b * P_ + t + 1];
    const float tc  = seq_times[b * P_ + t];
    const float npm = mask[b * P_ + t + 1];
    const float dtv = (tn - tc) * npm;

    const size_t rbase = (size_t)(b * PM1 + t) * DIN_;
    if (tid < DIN_) {
        sS[tid] = start_i[rbase + tid];
        sC[tid] = conv_i[rbase + tid];
        sO[tid] = omega_i[rbase + tid];
    }
    for (int i = tid; i < DIN_ * K_; i += 256) ldsw[i] = W_int[i];
    if (tid < S_) ltau[tid] = dtv * taus_u[(size_t)(b * PM1 + t) * S_ + tid];
    if (tid == 0) acc_blk = 0.0f;
    __syncthreads();

    for (int idx = tid; idx < S_ * CTS; idx += 256) {
        int s = idx / CTS;
        int d = idx - s * CTS;
        float v = 0.0f;
        if (d < DIN_) {
            float cv = sC[d];
            v = fast_tanh(cv + (sS[d] - cv) * __expf(-sO[d] * ltau[s]));
        }
        ct[idx] = v;
    }
    __syncthreads();

    if (wv < 7) {
        const int row  = lane & 15;
        const int koff = (lane >> 4) * 2;
        for (int tt = wv * 2; tt < wv * 2 + 2; ++tt) {
            int st = tt >> 1, kt = tt & 1;
            int s0 = st * 16, j0 = kt * 16;
            int  srow    = s0 + row;
            bool arow_ok = (srow < S_);
            int  jcol    = j0 + row;
            bool bcol_ok = (jcol < K_);
            v8f c = {};
            for (int k0 = 0; k0 < DIN_; k0 += 4) {
                int kk = k0 + koff;
                v2f a, bb;
                a.x  = arow_ok ? ct[srow * CTS + kk]     : 0.0f;
                a.y  = arow_ok ? ct[srow * CTS + kk + 1] : 0.0f;
                bb.x = bcol_ok ? ldsw[kk * K_ + jcol]       : 0.0f;
                bb.y = bcol_ok ? ldsw[(kk + 1) * K_ + jcol] : 0.0f;
                c = __builtin_amdgcn_wmma_f32_16x16x4_f32(
                        false, a, false, bb, (short)0, c, false, false);
            }
            float local = 0.0f;
            if (bcol_ok) {
                float bv = b_int[jcol];
                for (int r = 0; r < 8; ++r) {
                    int sr = s0 + r + ((lane >> 4) << 3);
                    if (sr < S_) local += softplus1(c[r] + bv);
                }
            }
            local = wave_sum(local);
            if (lane == 0) atomicAdd(&acc_blk, local);
        }
    } else {
        // log_sum term: only the seq_types-selected W_int column is needed
        int ty = seq_types[b * P_ + t + 1];
        int ksel = ty - 1;
        if (ksel < 0) ksel = 0;
        if (ksel >= K_) ksel = K_ - 1;
        float dot = 0.0f;
        for (int i = 0; i < 4; ++i) {
            int d = lane + 32 * i;
            float cv = sC[d];
            float ctd = fast_tanh(cv + (sS[d] - cv) * __expf(-sO[d] * dtv));
            dot += ctd * ldsw[d * K_ + ksel];
        }
        dot = wave_sum(dot);
        if (lane == 0) {
            float z = dot + b_int[ksel];
            float sumK = (npm > 0.0f) ? softplus1(z) : 1.0f;
            atomicAdd(&out[b * 2 + 0], logf(sumK));
        }
    }
    __syncthreads();
    if (tid == 0) {
        atomicAdd(&out[b * 2 + 1], dtv * npm * acc_blk * (1.0f / (float)S_));
    }
}

// ---------------------------------------------------------------------------
extern "C" void kernel_launch(void* const* d_in, const int* in_sizes, int n_in,
                              void* d_out, int out_size, void* d_ws, size_t ws_size,
                              hipStream_t stream)
{
    const float* values    = (const float*)d_in[0];
    const float* preatt    = (const float*)d_in[1];
    const float* mask      = (const float*)d_in[2];
    const float* seq_times = (const float*)d_in[3];
    const float* taus_u    = (const float*)d_in[4];
    const int*   seq_types = (const int*)d_in[5];
    const float* W_start   = (const float*)d_in[6];
    const float* b_start   = (const float*)d_in[7];
    const float* W_conv    = (const float*)d_in[8];
    const float* b_conv    = (const float*)d_in[9];
    const float* W_dec     = (const float*)d_in[10];
    const float* b_dec     = (const float*)d_in[11];
    const float* W_int     = (const float*)d_in[12];
    const float* b_int     = (const float*)d_in[13];
    float* out = (float*)d_out;

    float* embed   = (float*)d_ws;
    float* start_o = embed   + (size_t)B_ * P_   * DIN_;
    float* conv_o  = start_o + (size_t)B_ * PM1 * DIN_;
    float* omega_o = conv_o  + (size_t)B_ * PM1 * DIN_;

    athp_scan<<<B_ * M_, 256, 0, stream>>>(values, preatt, mask, embed, out);
    athp_mlp<<<B_ * 24, 256, 0, stream>>>(embed, mask,
                                          W_start, b_start, W_conv, b_conv,
                                          W_dec, b_dec,
                                          start_o, conv_o, omega_o);
    athp_tail<<<B_ * PM1, 256, 0, stream>>>(start_o, conv_o, omega_o,
                                            seq_times, mask, taus_u, seq_types,
                                            W_int, b_int, out);
}